// BertProber_10754598109583
// MI455X (gfx1250) — compile-verified
//
#include <hip/hip_runtime.h>
#include <stdint.h>

// Problem constants (match reference)
#define N_SENT 1024
#define SEQ_L  128
#define HID    768
#define KSPAN  5

typedef float v2f __attribute__((ext_vector_type(2)));
typedef float v8f __attribute__((ext_vector_type(8)));

// LDS layout: [ws 128 f32][wp 128 f32][tile buf0][tile buf1]
#define TILE_L      16
#define ROW_PAD     772                    // 768 + 4 floats: shifts banks between rows
#define TILE_FLOATS (TILE_L * ROW_PAD)     // 12352
#define TILE_BYTES  (TILE_FLOATS * 4)      // 49408
#define MASK_BYTES  (2 * SEQ_L * 4)        // 1024
#define LDS_BYTES   (MASK_BYTES + 2 * TILE_BYTES)  // 99840

// One async b128 copy: global (SGPR base + 32-bit voffset) -> LDS (per-lane addr).
// Tracked on ASYNCcnt; no VGPR data return.
__device__ __forceinline__ void async_b128(unsigned lds_off, unsigned goff,
                                           const float* sbase) {
  asm volatile("global_load_async_to_lds_b128 %0, %1, %2"
               :: "v"(lds_off), "v"(goff), "s"(sbase)
               : "memory");
}

// Issue the 12 async b128 loads this thread owns for one 16x768 tile.
__device__ __forceinline__ void issue_tile(unsigned buf_off, int tile,
                                           const float* sbase, int tid) {
#pragma unroll
  for (int i = 0; i < 12; ++i) {
    unsigned e   = (unsigned)(i * 256 + tid);   // 0..3071 float4 elements
    unsigned row = e / 192u;                    // 0..15
    unsigned c4  = e - row * 192u;              // 0..191
    unsigned lds_off = buf_off + (row * 193u + c4) * 16u;     // padded rows
    unsigned goff    = (unsigned)tile * (TILE_L * HID * 4u)
                     + row * (HID * 4u) + c4 * 16u;
    async_b128(lds_off, goff, sbase);
  }
}

__global__ __launch_bounds__(256)
void bert_prober_pool_kernel(const float* __restrict__ review_feat,
                             const float* __restrict__ reply_feat,
                             const int* __restrict__ rev_nt,
                             const int* __restrict__ rep_nt,
                             const int* __restrict__ rs, const int* __restrict__ re_,
                             const int* __restrict__ ps, const int* __restrict__ pe,
                             float* __restrict__ out) {
  extern __shared__ char smem[];
  float* ws = (float*)smem;            // sentence mask weights, [128]
  float* wp = ws + SEQ_L;              // span-union mask weights, [128]
  float* bufs = (float*)(smem + MASK_BYTES);
  const unsigned lds_base = (unsigned)(unsigned long long)(void*)smem;

  const int b = blockIdx.x;
  const int t = b >> 10;               // 0 = review, 1 = reply
  const int n = b & (N_SENT - 1);

  const float* feat = (t ? reply_feat : review_feat) + (size_t)n * SEQ_L * HID;
  const int nt = (t ? rep_nt : rev_nt)[n];
  const int* sst = (t ? ps : rs) + n * KSPAN;
  const int* sen = (t ? pe : re_) + n * KSPAN;

  int st[KSPAN], en[KSPAN];
  float cspan = 0.f;
#pragma unroll
  for (int j = 0; j < KSPAN; ++j) {
    st[j] = sst[j];
    en[j] = sen[j];
    int s = st[j] < 0 ? 0 : st[j];
    int e = en[j] > (SEQ_L - 1) ? (SEQ_L - 1) : en[j];
    if (e >= s) cspan += (float)(e - s + 1);   // spans are non-overlapping
  }

  const int tid   = threadIdx.x;
  const int lane  = tid & 31;
  const int wid   = tid >> 5;          // 8 waves
  const int laneM = lane & 15;         // A/D row (M) and B/D column (N) index
  const int hiK   = lane >> 4;         // hi half-lanes hold K+2 / K+3

  // Build per-position mask weights in LDS.
  if (tid < SEQ_L) {
    const int l = tid;
    float m = (l >= 1 && l <= nt) ? 1.f : 0.f;
    float sp = 0.f;
#pragma unroll
    for (int j = 0; j < KSPAN; ++j)
      if (l >= st[j] && l <= en[j]) sp = 1.f;
    ws[l] = m;
    wp[l] = sp;
  }

  // Accumulators: 6 H-chunks of 16 per wave (8*6*16 = 768).
  const v8f vzero = {0.f, 0.f, 0.f, 0.f, 0.f, 0.f, 0.f, 0.f};
  v8f acc[6];
#pragma unroll
  for (int c = 0; c < 6; ++c) acc[c] = vzero;

  // Prefetch tile 0, then double-buffered stream over 8 tiles of 16 rows.
  issue_tile(lds_base + MASK_BYTES, 0, feat, tid);
  __syncthreads();   // masks ready (also orders with first compute)

  for (int tle = 0; tle < 8; ++tle) {
    if (tle < 7) {
      issue_tile(lds_base + MASK_BYTES + (unsigned)((tle + 1) & 1) * TILE_BYTES,
                 tle + 1, feat, tid);
      asm volatile("s_wait_asynccnt 12" ::: "memory");  // tile tle done, tile tle+1 in flight
    } else {
      asm volatile("s_wait_asynccnt 0" ::: "memory");
    }
    __syncthreads();  // all waves' portions of this tile landed in LDS

    const float* buf = bufs + (tle & 1) * TILE_FLOATS;

#pragma unroll
    for (int c = 0; c < 6; ++c) {
      const int h0 = (wid * 6 + c) * 16 + laneM;
#pragma unroll
      for (int ks = 0; ks < 4; ++ks) {
        const int kl = ks * 4 + 2 * hiK;       // local K row pair for this half-wave
        const int kg = tle * TILE_L + kl;      // global K

        // A fragment (16x4 f32): row 0 = sent mask, row 1 = span mask, rest 0.
        const float s0 = ws[kg], s1 = ws[kg + 1];
        const float p0 = wp[kg], p1 = wp[kg + 1];
        v2f A;
        A.x = (laneM == 0) ? s0 : ((laneM == 1) ? p0 : 0.f);
        A.y = (laneM == 0) ? s1 : ((laneM == 1) ? p1 : 0.f);

        // B fragment (4x16 f32): rows kl, kl+1 of the LDS tile, cols h0..h0+15.
        v2f B;
        B.x = buf[kl * ROW_PAD + h0];
        B.y = buf[(kl + 1) * ROW_PAD + h0];

        acc[c] = __builtin_amdgcn_wmma_f32_16x16x4_f32(
            /*neg_a=*/false, A, /*neg_b=*/false, B,
            /*c_mod=*/(short)0, acc[c], /*reuse_a=*/false, /*reuse_b=*/false);
      }
    }
    __syncthreads();  // everyone done reading buf before it is refilled
  }

  // D layout: VGPR r, lanes 0-15 -> M=r, N=lane. Row 0 = sent dot, row 1 = span dot.
  const float inv_sent = 1.0f / (float)nt;
  const bool  has_span = cspan > 0.f;
  const float inv_span = 1.0f / fmaxf(cspan, 1.0f);

  float* pt_out   = out + (size_t)(2 * t)     * N_SENT * HID + (size_t)n * HID;
  float* sent_out = out + (size_t)(2 * t + 1) * N_SENT * HID + (size_t)n * HID;

  if (lane < 16) {
#pragma unroll
    for (int c = 0; c < 6; ++c) {
      const int h = (wid * 6 + c) * 16 + lane;
      const float sv = acc[c][0] * inv_sent;
      const float pv = has_span ? acc[c][1] * inv_span : sv;
      sent_out[h] = sv;
      pt_out[h]   = pv;
    }
  }
}

extern "C" void kernel_launch(void* const* d_in, const int* in_sizes, int n_in,
                              void* d_out, int out_size, void* d_ws, size_t ws_size,
                              hipStream_t stream) {
  (void)in_sizes; (void)n_in; (void)out_size; (void)d_ws; (void)ws_size;
  const float* review_feat = (const float*)d_in[0];
  const float* reply_feat  = (const float*)d_in[1];
  const int* rev_nt = (const int*)d_in[2];
  const int* rep_nt = (const int*)d_in[3];
  const int* rs = (const int*)d_in[4];
  const int* re_ = (const int*)d_in[5];
  const int* ps = (const int*)d_in[6];
  const int* pe = (const int*)d_in[7];
  float* out = (float*)d_out;

  dim3 grid(2 * N_SENT);   // (tensor, sentence) pairs
  dim3 block(256);         // 8 wave32s
  bert_prober_pool_kernel<<<grid, block, LDS_BYTES, stream>>>(
      review_feat, reply_feat, rev_nt, rep_nt, rs, re_, ps, pe, out);
}